// ContinuousConvEmbedding_49194555408680
// MI455X (gfx1250) — compile-verified
//
#include <hip/hip_runtime.h>
#include <hip/hip_bf16.h>

typedef __attribute__((ext_vector_type(2))) float v2f;
typedef __attribute__((ext_vector_type(8))) float v8f;

#define KS     3
#define K3     27
#define C_IN   8
#define KC     (K3 * C_IN)   // 216 contraction length (divisible by 4)
#define C_OUT  64
#define EPSF   1e-12f

// ---------------------------------------------------------------------------
// Phase 1: per-output-point scatter of trilinear corner weights into
// B[i, bin*8 + c].  One wave (32 lanes) per output point; 8 waves per block.
// Accumulation in LDS via ds_add_f32 atomics, then streamed to workspace.
// ---------------------------------------------------------------------------
__global__ __launch_bounds__(256) void cconv_scatter_kernel(
    const float* __restrict__ feats,    // [Ni, 8]
    const float* __restrict__ pos_in,   // [Ni, 3]
    const float* __restrict__ pos_out,  // [No, 3]
    const float* __restrict__ extents,  // [1]
    float*       __restrict__ Bg,       // [No, 216] workspace
    int Ni, int No)
{
    __shared__ float Bsh[8][KC];        // 8 output points per block, 6.9 KB

    const int lane = threadIdx.x & 31;
    const int wv   = threadIdx.x >> 5;
    const int i    = blockIdx.x * 8 + wv;   // output point owned by this wave

    // zero this wave's accumulator tile
    for (int t = lane; t < KC; t += 32) Bsh[wv][t] = 0.0f;
    __syncthreads();

    const float inv = 2.0f / extents[0];

    if (i < No) {
        const float ox = pos_out[i * 3 + 0];
        const float oy = pos_out[i * 3 + 1];
        const float oz = pos_out[i * 3 + 2];

        for (int j = lane; j < Ni; j += 32) {
            const float rx = (pos_in[j * 3 + 0] - ox) * inv;
            const float ry = (pos_in[j * 3 + 1] - oy) * inv;
            const float rz = (pos_in[j * 3 + 2] - oz) * inv;
            const float r2 = rx * rx + ry * ry + rz * rz;
            if (r2 >= 1.0f) continue;            // window == 0 -> no contribution

            const float om     = 1.0f - r2;
            const float window = om * om * om;   // poly6
            const float r      = sqrtf(fmaxf(r2, EPSF));
            const float linf   = fmaxf(fabsf(rx), fmaxf(fabsf(ry), fabsf(rz)));
            const float scale  = (linf > EPSF) ? (r / fmaxf(linf, EPSF)) : 1.0f;

            // ball -> cube, then grid coords t = (q+1)*0.5*(KS-1) = q+1
            const float t0f = fminf(fmaxf(rx * scale, -1.0f), 1.0f) + 1.0f;
            const float t1f = fminf(fmaxf(ry * scale, -1.0f), 1.0f) + 1.0f;
            const float t2f = fminf(fmaxf(rz * scale, -1.0f), 1.0f) + 1.0f;

            int i0 = (int)floorf(t0f); i0 = i0 < 0 ? 0 : (i0 > 1 ? 1 : i0);
            int i1 = (int)floorf(t1f); i1 = i1 < 0 ? 0 : (i1 > 1 ? 1 : i1);
            int i2 = (int)floorf(t2f); i2 = i2 < 0 ? 0 : (i2 > 1 ? 1 : i2);

            const float f0 = t0f - (float)i0;
            const float f1 = t1f - (float)i1;
            const float f2 = t2f - (float)i2;

            // features of input j (8 channels, two 16B loads)
            const float4* fp = (const float4*)(feats + (size_t)j * C_IN);
            const float4  fa = fp[0];
            const float4  fb = fp[1];

            #pragma unroll
            for (int o0 = 0; o0 < 2; ++o0) {
                const float w0 = (o0 ? f0 : 1.0f - f0) * window;
                #pragma unroll
                for (int o1 = 0; o1 < 2; ++o1) {
                    const float w01 = w0 * (o1 ? f1 : 1.0f - f1);
                    #pragma unroll
                    for (int o2 = 0; o2 < 2; ++o2) {
                        const float w = w01 * (o2 ? f2 : 1.0f - f2);
                        const int bin = (i0 + o0) * 9 + (i1 + o1) * 3 + (i2 + o2);
                        float* dst = &Bsh[wv][bin * C_IN];
                        atomicAdd(dst + 0, w * fa.x);
                        atomicAdd(dst + 1, w * fa.y);
                        atomicAdd(dst + 2, w * fa.z);
                        atomicAdd(dst + 3, w * fa.w);
                        atomicAdd(dst + 4, w * fb.x);
                        atomicAdd(dst + 5, w * fb.y);
                        atomicAdd(dst + 6, w * fb.z);
                        atomicAdd(dst + 7, w * fb.w);
                    }
                }
            }
        }
    }
    __syncthreads();

    if (i < No) {
        for (int t = lane; t < KC; t += 32)
            Bg[(size_t)i * KC + t] = Bsh[wv][t];
    }
}

// ---------------------------------------------------------------------------
// Phase 2: out[No,64] = relu(B[No,216] @ K[216,64] + bias) via fp32 WMMA.
// One wave per 16x16 output tile; K loop of 54 x v_wmma_f32_16x16x4_f32.
// Lane layout (wave32):
//   A (16x4):  lanes 0-15 hold row M=lane, K = {k, k+1};
//              lanes 16-31 hold row M=lane-16, K = {k+2, k+3}
//   B (4x16):  lanes 0-15 hold col N=lane, K = {k, k+1};
//              lanes 16-31 hold col N=lane-16, K = {k+2, k+3}
//   C/D:       VGPR v: lanes 0-15 -> M=v, N=lane; lanes 16-31 -> M=v+8
// ---------------------------------------------------------------------------
__global__ __launch_bounds__(256) void cconv_gemm_wmma_kernel(
    const float* __restrict__ Bg,    // [No, 216]
    const float* __restrict__ Wk,    // [216, 64] (= kernel [27,8,64] flattened)
    const float* __restrict__ bias,  // [64]
    float*       __restrict__ out,   // [No, 64]
    int No)
{
    const int lane   = threadIdx.x & 31;
    const int wv     = threadIdx.x >> 5;
    const int tile   = blockIdx.x * 8 + wv;
    const int ntileC = C_OUT / 16;           // 4 column tiles
    const int tr     = tile / ntileC;
    const int tc     = tile - tr * ntileC;
    const int row0   = tr * 16;
    const int col0   = tc * 16;
    if (row0 >= No) return;

    const int half = lane >> 4;              // 0: lanes 0-15, 1: lanes 16-31
    const int l16  = lane & 15;

    // A-fragment row for this lane (clamped for ragged tails)
    int arow = row0 + l16;
    if (arow >= No) arow = No - 1;
    const float* arow_p = Bg + (size_t)arow * KC;

    v8f acc = {};
    #pragma unroll 2
    for (int k = 0; k < KC; k += 4) {
        const int ka = k + half * 2;         // lanes 0-15: k,k+1 ; 16-31: k+2,k+3
        v2f a;
        a.x = arow_p[ka + 0];
        a.y = arow_p[ka + 1];
        v2f b;
        b.x = Wk[(size_t)(ka + 0) * C_OUT + col0 + l16];
        b.y = Wk[(size_t)(ka + 1) * C_OUT + col0 + l16];
        acc = __builtin_amdgcn_wmma_f32_16x16x4_f32(
            /*neg_a=*/false, a, /*neg_b=*/false, b,
            /*c_mod=*/(short)0, acc, /*reuse_a=*/false, /*reuse_b=*/false);
    }

    const int   col = col0 + l16;
    const float bv  = bias[col];
    float* outp = out + (size_t)row0 * C_OUT + (size_t)half * 8 * C_OUT + col;

    if (row0 + 16 <= No) {
        // Fast path: full 16x16 tile, unconditional coalesced stores.
        #pragma unroll
        for (int v = 0; v < 8; ++v)
            outp[(size_t)v * C_OUT] = fmaxf(acc[v] + bv, 0.0f);
    } else {
        // Ragged tail (not hit for No % 16 == 0).
        #pragma unroll
        for (int v = 0; v < 8; ++v) {
            const int row = row0 + v + half * 8;
            if (row < No)
                out[(size_t)row * C_OUT + col] = fmaxf(acc[v] + bv, 0.0f);
        }
    }
}

// ---------------------------------------------------------------------------
// Host-side launcher
// ---------------------------------------------------------------------------
extern "C" void kernel_launch(void* const* d_in, const int* in_sizes, int n_in,
                              void* d_out, int out_size, void* d_ws, size_t ws_size,
                              hipStream_t stream) {
    const float* feats   = (const float*)d_in[0];   // [Ni, 8]
    const float* pos_in  = (const float*)d_in[1];   // [Ni, 3]
    const float* pos_out = (const float*)d_in[2];   // [No, 3]
    const float* extents = (const float*)d_in[3];   // [1]
    const float* kern    = (const float*)d_in[4];   // [27, 8, 64] -> [216, 64]
    const float* bias    = (const float*)d_in[5];   // [64]

    const int Ni = in_sizes[0] / C_IN;
    const int No = in_sizes[2] / 3;

    float* Bg = (float*)d_ws;                       // [No, 216] fp32 scratch

    // Phase 1: pairwise scatter into B
    const int nblk1 = (No + 7) / 8;                 // 8 output points / block
    cconv_scatter_kernel<<<nblk1, 256, 0, stream>>>(
        feats, pos_in, pos_out, extents, Bg, Ni, No);

    // Phase 2: WMMA GEMM + bias + ReLU
    const int rowTiles = (No + 15) / 16;
    const int tiles    = rowTiles * (C_OUT / 16);
    const int nblk2    = (tiles + 7) / 8;           // 8 waves / block
    cconv_gemm_wmma_kernel<<<nblk2, 256, 0, stream>>>(
        Bg, kern, bias, (float*)d_out, No);
}